// GCN_7851200217412
// MI455X (gfx1250) — compile-verified
//
#include <hip/hip_runtime.h>
#include <math.h>

typedef __attribute__((ext_vector_type(2))) float v2f;
typedef __attribute__((ext_vector_type(8))) float v8f;

#define NNODES 50000
#define BN_EPS 1e-5f

// ---------------- degree / dinv ----------------
__global__ void k_fill1(float* deg, int n) {
    int i = blockIdx.x * blockDim.x + threadIdx.x;
    if (i < n) deg[i] = 1.0f;   // self-loop contributes 1 to in-degree
}

__global__ void k_deg(const int* __restrict__ dst, float* __restrict__ deg, int E) {
    int e = blockIdx.x * blockDim.x + threadIdx.x;
    if (e < E) atomicAdd(&deg[dst[e]], 1.0f);
}

__global__ void k_rsqrt_inplace(float* d, int n) {
    int i = blockIdx.x * blockDim.x + threadIdx.x;
    if (i < n) d[i] = rsqrtf(d[i]);   // deg >= 1 always
}

// ---------------- FP32 WMMA GEMM: H[M x N] = X[M x K] @ W[K x N] ----------------
// Block = 128 threads = 4 waves; block computes one 16-row M tile across full N.
// Each wave owns TILES consecutive 16-col tiles. N = 64 * TILES.
template <int K, int TILES>
__global__ __launch_bounds__(128) void k_gemm_wmma(const float* __restrict__ X,
                                                   const float* __restrict__ W,
                                                   float* __restrict__ H, int M) {
    constexpr int N = 64 * TILES;
    __shared__ float As[16][K + 4];              // +4 pad: avoid 16-way bank conflict
    const int m0  = blockIdx.x * 16;
    const int tid = threadIdx.x;

    // Stage 16 x K A-tile into LDS (fully coalesced: global index m0*K + i)
    for (int i = tid; i < 16 * K; i += 128)
        As[i / K][i % K] = X[(size_t)m0 * K + i];
    __syncthreads();

    const int lane = tid & 31;
    const int wave = tid >> 5;
    const int hv   = lane >> 4;    // half-wave selector
    const int lr   = lane & 15;
    const int nb   = wave * (TILES * 16);

    v8f acc[TILES];
#pragma unroll
    for (int t = 0; t < TILES; ++t)
#pragma unroll
        for (int r = 0; r < 8; ++r) acc[t][r] = 0.0f;

    for (int k = 0; k < K; k += 4) {
        // A fragment 16x4: VGPR v holds A[M=lr][K = k + 2*hv + v]
        v2f a;
        a[0] = As[lr][k + 2 * hv];
        a[1] = As[lr][k + 2 * hv + 1];
#pragma unroll
        for (int t = 0; t < TILES; ++t) {
            int n = nb + t * 16 + lr;
            // B fragment 4x16: VGPR v holds B[K = k + 2*hv + v][N = lr]
            v2f b;
            b[0] = W[(size_t)(k + 2 * hv) * N + n];
            b[1] = W[(size_t)(k + 2 * hv + 1) * N + n];
            acc[t] = __builtin_amdgcn_wmma_f32_16x16x4_f32(
                false, a, false, b, (short)0, acc[t], false, false);
        }
    }

#pragma unroll
    for (int t = 0; t < TILES; ++t)
#pragma unroll
        for (int r = 0; r < 8; ++r) {
            int row = m0 + r + 8 * hv;          // C/D: VGPR r -> M = r + 8*half
            int col = nb + t * 16 + lr;         // N = lane % 16
            H[(size_t)row * N + col] = acc[t][r];
        }
}

// ---------------- out = h * dinv^2 + bias  (self-loop term + bias init) ----------------
__global__ void k_self_bias(const float* __restrict__ h, const float* __restrict__ dinv,
                            const float* __restrict__ bias, float* __restrict__ out,
                            int n, int C) {
    long long i = (long long)blockIdx.x * blockDim.x + threadIdx.x;
    if (i < (long long)n * C) {
        int node = (int)(i / C), c = (int)(i % C);
        float di = dinv[node];
        out[i] = h[i] * di * di + bias[c];
    }
}

// ---------------- edge aggregation: out[dst] += h[src] * dinv[src]*dinv[dst] ----------------
// One wave per edge (grid-stride); lanes split channels with float4 gathers + f32 atomics.
__global__ void k_edge_agg(const float* __restrict__ h, const int* __restrict__ src,
                           const int* __restrict__ dst, const float* __restrict__ dinv,
                           float* __restrict__ out, int E, int C) {
    int gw   = (blockIdx.x * blockDim.x + threadIdx.x) >> 5;
    int nw   = (gridDim.x * blockDim.x) >> 5;
    int lane = threadIdx.x & 31;
    for (int e = gw; e < E; e += nw) {
        int s = src[e], d = dst[e];
        float nrm = dinv[s] * dinv[d];
        const float4* hs = reinterpret_cast<const float4*>(h + (size_t)s * C);
        float* od = out + (size_t)d * C;
        for (int c4 = lane; c4 * 4 < C; c4 += 32) {
            float4 v = hs[c4];
            atomicAdd(&od[c4 * 4 + 0], v.x * nrm);
            atomicAdd(&od[c4 * 4 + 1], v.y * nrm);
            atomicAdd(&od[c4 * 4 + 2], v.z * nrm);
            atomicAdd(&od[c4 * 4 + 3], v.w * nrm);
        }
    }
}

// ---------------- batch norm ----------------
__global__ void k_zero(float* p, int n) {
    int i = blockIdx.x * blockDim.x + threadIdx.x;
    if (i < n) p[i] = 0.0f;
}

// blockDim.x == C; thread c accumulates sum/sumsq over a row range (coalesced rows)
__global__ void k_bn_stats(const float* __restrict__ x, float* __restrict__ stats,
                           int M, int C, int rows_per) {
    int c  = threadIdx.x;
    int r0 = blockIdx.x * rows_per;
    int r1 = min(M, r0 + rows_per);
    float s = 0.0f, ss = 0.0f;
    for (int r = r0; r < r1; ++r) {
        float v = x[(size_t)r * C + c];
        s += v; ss += v * v;
    }
    atomicAdd(&stats[c], s);
    atomicAdd(&stats[C + c], ss);
}

__global__ void k_bn_relu(const float* __restrict__ x, const float* __restrict__ stats,
                          const float* __restrict__ gamma, const float* __restrict__ beta,
                          float* __restrict__ y, int M, int C) {
    long long i = (long long)blockIdx.x * blockDim.x + threadIdx.x;
    if (i < (long long)M * C) {
        int c = (int)(i % C);
        float mean = stats[c] / (float)M;
        float var  = stats[C + c] / (float)M - mean * mean;
        float v = (x[i] - mean) * rsqrtf(var + BN_EPS) * gamma[c] + beta[c];
        y[i] = v > 0.0f ? v : 0.0f;
    }
}

// ---------------- launch ----------------
extern "C" void kernel_launch(void* const* d_in, const int* in_sizes, int n_in,
                              void* d_out, int out_size, void* d_ws, size_t ws_size,
                              hipStream_t stream) {
    const float* x0    = (const float*)d_in[0];   // node_feat [50000 x 128]
    const int*   ei    = (const int*)d_in[1];     // edge_index [2 x E]
    const float* W1    = (const float*)d_in[2];   // [128 x 256]
    const float* b1    = (const float*)d_in[3];
    const float* gamma = (const float*)d_in[4];
    const float* beta  = (const float*)d_in[5];
    const float* W2    = (const float*)d_in[6];   // [256 x 128]
    const float* b2    = (const float*)d_in[7];

    const int Nn  = NNODES;
    const int E   = in_sizes[1] / 2;
    const int CH  = 256, COUT = 128;
    const int* src = ei;
    const int* dst = ei + E;

    float* ws    = (float*)d_ws;
    float* dinv  = ws;                              // 50000
    float* stats = ws + 50000;                      // 512
    float* h1    = ws + 50512;                      // 50000*256 (GEMM1 out, later X2)
    float* agg   = h1 + (size_t)Nn * CH;            // 50000*256 (agg1, later H2)
    float* out   = (float*)d_out;                   // 50000*128

    // degrees -> dinv (in place)
    k_fill1<<<(Nn + 255) / 256, 256, 0, stream>>>(dinv, Nn);
    k_deg<<<(E + 255) / 256, 256, 0, stream>>>(dst, dinv, E);
    k_rsqrt_inplace<<<(Nn + 255) / 256, 256, 0, stream>>>(dinv, Nn);

    // layer 1: H1 = X @ W1   (K=128, N=256)
    k_gemm_wmma<128, 4><<<Nn / 16, 128, 0, stream>>>(x0, W1, h1, Nn);
    long long tot1 = (long long)Nn * CH;
    k_self_bias<<<(int)((tot1 + 255) / 256), 256, 0, stream>>>(h1, dinv, b1, agg, Nn, CH);
    k_edge_agg<<<2048, 256, 0, stream>>>(h1, src, dst, dinv, agg, E, CH);

    // batch norm + relu (X2 written over h1)
    k_zero<<<2, 256, 0, stream>>>(stats, 2 * CH);
    int rows_per = (Nn + 127) / 128;
    k_bn_stats<<<128, CH, 0, stream>>>(agg, stats, Nn, CH, rows_per);
    k_bn_relu<<<(int)((tot1 + 255) / 256), 256, 0, stream>>>(agg, stats, gamma, beta, h1, Nn, CH);

    // layer 2: H2 = X2 @ W2   (K=256, N=128), H2 into agg buffer
    k_gemm_wmma<256, 2><<<Nn / 16, 128, 0, stream>>>(h1, W2, agg, Nn);
    long long tot2 = (long long)Nn * COUT;
    k_self_bias<<<(int)((tot2 + 255) / 256), 256, 0, stream>>>(agg, dinv, b2, out, Nn, COUT);
    k_edge_agg<<<2048, 256, 0, stream>>>(agg, src, dst, dinv, out, E, COUT);
}